// HeteroGraphSage_82420422410369
// MI455X (gfx1250) — compile-verified
//
#include <hip/hip_runtime.h>

#define N_NODES 100000
#define DIM 64
#define N_EDGES 1600000

typedef __attribute__((ext_vector_type(2))) float v2f;
typedef __attribute__((ext_vector_type(8))) float v8f;

// ---------------------------------------------------------------------------
// Zero a float region of workspace (graph-capture-safe, no hipMemset).
// ---------------------------------------------------------------------------
__global__ void zero_kernel(float* __restrict__ p, int n) {
  int i = blockIdx.x * blockDim.x + threadIdx.x;
  if (i < n) p[i] = 0.0f;
}

// ---------------------------------------------------------------------------
// Degree count: one thread per edge, atomic add 1.0 at dst.
// Degrees are identical for both layers -> computed once.
// ---------------------------------------------------------------------------
__global__ void degree_kernel(const long long* __restrict__ dst,
                              float* __restrict__ cnt, int e) {
  int i = blockIdx.x * blockDim.x + threadIdx.x;
  if (i < e) atomicAdd(&cnt[(int)dst[i]], 1.0f);
}

// ---------------------------------------------------------------------------
// Edge scatter: 16 threads per edge, each thread moves a float4 chunk
// (coalesced 256B row gather) and does 4 f32 atomic adds into agg, which is
// 25.6MB and L2-resident on MI455X (192MB L2).
// ---------------------------------------------------------------------------
__global__ void scatter_kernel(const float* __restrict__ h,
                               const long long* __restrict__ src,
                               const long long* __restrict__ dst,
                               float* __restrict__ agg, int e) {
  long long t = (long long)blockIdx.x * blockDim.x + threadIdx.x;
  long long ed = t >> 4;
  if (ed >= e) return;
  int c = ((int)t & 15) << 2;
  int s = (int)src[ed];
  int d = (int)dst[ed];
  const float4 v = *(const float4*)(h + (size_t)s * DIM + c);
  float* p = agg + (size_t)d * DIM + c;
  atomicAdd(p + 0, v.x);
  atomicAdd(p + 1, v.y);
  atomicAdd(p + 2, v.z);
  atomicAdd(p + 3, v.w);
}

// ---------------------------------------------------------------------------
// Fused SAGE layer GEMM using V_WMMA_F32_16X16X4_F32 (full fp32 precision).
//   out = prelu(agg*inv @ wl.T + h @ wr.T + bl, alpha) + x0 @ wsk.T + bsk
// Block = 256 threads = 8 wave32 -> 32 rows x 64 cols per block.
// Wave w: rows (w/4)*16.. , cols (w%4)*16.. (one 16x16 D-tile per wave).
//
// A fragment (16x4 f32, ISA layout): lanes 0-15 hold K = k+0,k+1; lanes 16-31
// hold K = k+2,k+3 for row M = lane&15  -> one contiguous float2 load.
// B fragment (4x16):  B[k'][j] = W[j*64+k'] (since out = A @ W.T), lane half
// selects K-pair like A -> one contiguous float2 load from W row j.
// ---------------------------------------------------------------------------
__global__ __launch_bounds__(256) void sage_gemm_kernel(
    const float* __restrict__ agg, const float* __restrict__ cnt,
    const float* __restrict__ h, const float* __restrict__ x0,
    const float* __restrict__ wl, const float* __restrict__ bl,
    const float* __restrict__ wr, const float* __restrict__ wsk,
    const float* __restrict__ bsk, const float* __restrict__ alpha,
    float* __restrict__ out) {
  const int lane = threadIdx.x & 31;
  const int wave = threadIdx.x >> 5;
  const int hi = lane >> 4;   // lane half (selects K+2/K+3)
  const int l15 = lane & 15;
  const int rowbase = blockIdx.x * 32 + (wave >> 2) * 16;
  const int col = (wave & 3) * 16 + l15;  // output column 0..63

  int aRow = rowbase + l15;
  if (aRow >= N_NODES) aRow = N_NODES - 1;  // clamp: EXEC must stay all-ones
  const int koff = hi * 2;
  const size_t arow64 = (size_t)aRow * DIM;
  const size_t wrow64 = (size_t)col * DIM;

  const float inv = 1.0f / fmaxf(cnt[aRow], 1.0f);

  v8f acc1 = {};  // agg@wl.T + h@wr.T   (pre-PReLU)
  v8f acc2 = {};  // x0@wsk.T            (skip path)
#pragma unroll
  for (int k = 0; k < DIM; k += 4) {
    const int kk = k + koff;
    v2f aAgg = *(const v2f*)(agg + arow64 + kk);
    aAgg *= inv;  // fold mean normalization into A fragment
    v2f bW = *(const v2f*)(wl + wrow64 + kk);
    acc1 = __builtin_amdgcn_wmma_f32_16x16x4_f32(false, aAgg, false, bW,
                                                 (short)0, acc1, false, false);
    v2f aH = *(const v2f*)(h + arow64 + kk);
    bW = *(const v2f*)(wr + wrow64 + kk);
    acc1 = __builtin_amdgcn_wmma_f32_16x16x4_f32(false, aH, false, bW,
                                                 (short)0, acc1, false, false);
    v2f aX = *(const v2f*)(x0 + arow64 + kk);
    bW = *(const v2f*)(wsk + wrow64 + kk);
    acc2 = __builtin_amdgcn_wmma_f32_16x16x4_f32(false, aX, false, bW,
                                                 (short)0, acc2, false, false);
  }

  // Epilogue: bias -> PReLU -> skip add. C/D layout: VGPR r holds row
  // (rowbase + r + 8*hi), column `col`.
  const float bl_c = bl[col];
  const float a_c = alpha[col];
  const float bs_c = bsk[col];
#pragma unroll
  for (int r = 0; r < 8; ++r) {
    const int row = rowbase + r + hi * 8;
    float v = acc1[r] + bl_c;
    v = (v >= 0.0f) ? v : a_c * v;
    v += acc2[r] + bs_c;
    if (row < N_NODES) out[(size_t)row * DIM + col] = v;
  }
}

// ---------------------------------------------------------------------------
// Host-side launch sequence (all on `stream`, graph-capture-safe).
// ws layout (floats): [0, N) cnt | [102400, +N*64) agg | next N*64 h1
// ---------------------------------------------------------------------------
extern "C" void kernel_launch(void* const* d_in, const int* in_sizes, int n_in,
                              void* d_out, int out_size, void* d_ws,
                              size_t ws_size, hipStream_t stream) {
  (void)in_sizes; (void)n_in; (void)out_size; (void)ws_size;
  const float* x    = (const float*)d_in[0];
  const long long* ei = (const long long*)d_in[1];  // [2, E] int64
  const float* wl0  = (const float*)d_in[2];
  const float* bl0  = (const float*)d_in[3];
  const float* wr0  = (const float*)d_in[4];
  const float* wsk0 = (const float*)d_in[5];
  const float* bsk0 = (const float*)d_in[6];
  const float* a0   = (const float*)d_in[7];
  const float* wl1  = (const float*)d_in[8];
  const float* bl1  = (const float*)d_in[9];
  const float* wr1  = (const float*)d_in[10];
  const float* wsk1 = (const float*)d_in[11];
  const float* bsk1 = (const float*)d_in[12];
  const float* a1   = (const float*)d_in[13];

  const long long* src = ei;
  const long long* dst = ei + N_EDGES;

  float* ws  = (float*)d_ws;
  float* cnt = ws;                               // N floats (pad to 102400)
  float* agg = ws + 102400;                      // N*64 floats
  float* h1  = agg + (size_t)N_NODES * DIM;      // N*64 floats

  const int TB = 256;
  const int nz0 = 102400 + N_NODES * DIM;        // cnt + agg together
  const int na = N_NODES * DIM;
  const long long stt = (long long)N_EDGES * 16; // scatter threads
  const unsigned sgrid = (unsigned)((stt + TB - 1) / TB);
  const int ggrid = (N_NODES + 31) / 32;

  // ---- shared preamble: zero cnt+agg, count degrees once ----
  zero_kernel<<<(nz0 + TB - 1) / TB, TB, 0, stream>>>(cnt, nz0);
  degree_kernel<<<(N_EDGES + TB - 1) / TB, TB, 0, stream>>>(dst, cnt, N_EDGES);

  // ---- layer 0 ----
  scatter_kernel<<<sgrid, TB, 0, stream>>>(x, src, dst, agg, N_EDGES);
  sage_gemm_kernel<<<ggrid, TB, 0, stream>>>(agg, cnt, x, x, wl0, bl0, wr0,
                                             wsk0, bsk0, a0, h1);

  // ---- layer 1 ----
  zero_kernel<<<(na + TB - 1) / TB, TB, 0, stream>>>(agg, na);
  scatter_kernel<<<sgrid, TB, 0, stream>>>(h1, src, dst, agg, N_EDGES);
  sage_gemm_kernel<<<ggrid, TB, 0, stream>>>(agg, cnt, h1, x, wl1, bl1, wr1,
                                             wsk1, bsk1, a1, (float*)d_out);
}